// FourierConv2D_35218731827819
// MI455X (gfx1250) — compile-verified
//
#include <hip/hip_runtime.h>
#include <hip/hip_bf16.h>

// FourierConv2D on gfx1250 (MI455X).
// Key insight: the reference crops F2 to a single frequency column (kx=127)
// and 255 rows (ky=127..381), so we never materialize the second 2D FFT.
// Pipeline:
//   F1  = (1/509) * 2D-DFT of zero-padded im          -> two WMMA matmul stages
//   T   = sum_{cin,x} Re(F1*W) * e^{-i*theta*127*x}   -> HBM-roofline pass over W
//   out = (1/509) * sum_y T * e^{-i*theta*ky*y} + b   -> tiny final DFT
// theta = 2*pi/509. Twiddles via exact (p mod 509) table for accuracy.

typedef float v2f __attribute__((ext_vector_type(2)));
typedef float v8f __attribute__((ext_vector_type(8)));

__device__ __forceinline__ v8f fc_wmma4(v2f a, v2f b, v8f c) {
  // V_WMMA_F32_16X16X4_F32 : D(16x16 f32) = A(16x4) * B(4x16) + C
  return __builtin_amdgcn_wmma_f32_16x16x4_f32(false, a, false, b, (short)0, c,
                                               false, false);
}

// ---------------- init kernels ------------------------------------------

__global__ void fc_init_tw(float* twc, float* tws) {
  int p = blockIdx.x * blockDim.x + threadIdx.x;
  if (p >= 512) return;
  if (p < 509) {
    double ang = 2.0 * 3.14159265358979323846 * (double)p / 509.0;
    twc[p] = (float)cos(ang);
    tws[p] = (float)sin(ang);
  } else {
    twc[p] = 0.0f;
    tws[p] = 0.0f;
  }
}

// Fill E1 (row DFT B-matrix, 256x256 padded) and column-DFT A-matrices
// cosT/sinP/sinN (512x256 padded). Zero padding rows/cols makes the WMMA
// K-loops guard-free.
__global__ void fc_init_E(const float* __restrict__ twc,
                          const float* __restrict__ tws,
                          float* __restrict__ E1r, float* __restrict__ E1i,
                          float* __restrict__ cosT, float* __restrict__ sinP,
                          float* __restrict__ sinN) {
  int i = blockIdx.x * blockDim.x + threadIdx.x;  // 512*256 threads
  if (i >= 512 * 256) return;
  int col = i & 255;   // v (row-DFT) or h (col-DFT)
  int row = i >> 8;    // w (row-DFT) or u (col-DFT)
  if (row < 256) {     // E1[w][v] = e^{-i*theta*v*w}
    float er = 0.0f, ei = 0.0f;
    if (row < 255 && col < 255) {
      int p = (row * col) % 509;
      er = twc[p];
      ei = -tws[p];
    }
    E1r[i] = er;
    E1i[i] = ei;
  }
  float c = 0.0f, s = 0.0f;  // e^{-i*theta*u*h} = cos - i*sin
  if (row < 509 && col < 255) {
    int p = (row * col) % 509;
    c = twc[p];
    s = tws[p];
  }
  cosT[i] = c;
  sinP[i] = s;
  sinN[i] = -s;
}

// Pack im (2,8,255,255) into padded [4096][256] (zero col 255, zero tail
// rows) and zero the G pad rows read by stage B's zero-coefficient K=255 tap.
__global__ void fc_pack_im(const float* __restrict__ im,
                           float* __restrict__ imP, float* __restrict__ Gr,
                           float* __restrict__ Gi) {
  int i = blockIdx.x * blockDim.x + threadIdx.x;  // 4096*256 threads
  if (i >= 4096 * 256) return;
  int col = i & 255;
  int row = i >> 8;
  float v = 0.0f;
  if (row < 4080 && col < 255) v = im[row * 255 + col];
  imP[i] = v;
  if (row >= 4080) {
    Gr[i] = 0.0f;
    Gi[i] = 0.0f;
  }
}

// ---------------- stage A: row DFT  G = imP x E1 ------------------------
// One wave per 16x16 output tile. M=4080 (255 tiles), N=256 (16 tiles), K=256.

__global__ void __launch_bounds__(256) fc_stageA(
    const float* __restrict__ imP, const float* __restrict__ E1r,
    const float* __restrict__ E1i, float* __restrict__ Gr,
    float* __restrict__ Gi) {
  int wave = (blockIdx.x * blockDim.x + threadIdx.x) >> 5;  // 0..4079
  int lane = threadIdx.x & 31;
  int mt = wave >> 4;  // 0..254
  int nt = wave & 15;  // 0..15
  int m0 = mt * 16, n0 = nt * 16;
  int half = lane >> 4, mrow = lane & 15;

  v8f cr = {};
  v8f ci = {};
  const float* arow = imP + (size_t)(m0 + mrow) * 256;
  for (int k0 = 0; k0 < 256; k0 += 4) {
    int ka = k0 + 2 * half;
    v2f a = *(const v2f*)(arow + ka);  // A[m, ka..ka+1], 8B aligned
    v2f br, bi;
    br.x = E1r[ka * 256 + n0 + mrow];
    br.y = E1r[(ka + 1) * 256 + n0 + mrow];
    bi.x = E1i[ka * 256 + n0 + mrow];
    bi.y = E1i[(ka + 1) * 256 + n0 + mrow];
    cr = fc_wmma4(a, br, cr);  // Gr += im * cos
    ci = fc_wmma4(a, bi, ci);  // Gi += im * (-sin)
  }
#pragma unroll
  for (int r = 0; r < 8; ++r) {
    int row = m0 + r + 8 * half;
    Gr[(size_t)row * 256 + n0 + mrow] = cr[r];
    Gi[(size_t)row * 256 + n0 + mrow] = ci[r];
  }
}

// ---------------- stage B: column DFT  F1 = E2^T x G (complex) ----------
// 16 images x (M=512 -> 32 tiles) x (N=256 -> 16 tiles), K=256; 4 WMMAs/step.

__global__ void __launch_bounds__(256) fc_stageB(
    const float* __restrict__ cosT, const float* __restrict__ sinP,
    const float* __restrict__ sinN, const float* __restrict__ Gr,
    const float* __restrict__ Gi, float* __restrict__ F1r,
    float* __restrict__ F1i) {
  int wave = (blockIdx.x * blockDim.x + threadIdx.x) >> 5;  // 0..8191
  int lane = threadIdx.x & 31;
  int img = wave >> 9;         // 0..15  (b*8+cin)
  int mt = (wave >> 4) & 31;   // u tile
  int nt = wave & 15;          // v tile
  int m0 = mt * 16, n0 = nt * 16;
  int half = lane >> 4, mrow = lane & 15;

  const float* gr = Gr + (size_t)img * 255 * 256;  // h rows contiguous
  const float* gi = Gi + (size_t)img * 255 * 256;
  const float* ac = cosT + (size_t)(m0 + mrow) * 256;
  const float* ap = sinP + (size_t)(m0 + mrow) * 256;
  const float* an = sinN + (size_t)(m0 + mrow) * 256;

  v8f fr = {};
  v8f fi = {};
  for (int k0 = 0; k0 < 256; k0 += 4) {
    int ka = k0 + 2 * half;
    v2f Ac = *(const v2f*)(ac + ka);
    v2f Ap = *(const v2f*)(ap + ka);
    v2f An = *(const v2f*)(an + ka);
    v2f Br, Bi;
    Br.x = gr[ka * 256 + n0 + mrow];
    Br.y = gr[(ka + 1) * 256 + n0 + mrow];
    Bi.x = gi[ka * 256 + n0 + mrow];
    Bi.y = gi[(ka + 1) * 256 + n0 + mrow];
    fr = fc_wmma4(Ac, Br, fr);  // += cos*Gr
    fr = fc_wmma4(Ap, Bi, fr);  // += sin*Gi
    fi = fc_wmma4(Ac, Bi, fi);  // += cos*Gi
    fi = fc_wmma4(An, Br, fi);  // += (-sin)*Gr
  }
  const float inv = 1.0f / 509.0f;  // ortho norm of first rfft2
  float* f1r = F1r + (size_t)img * 512 * 256;
  float* f1i = F1i + (size_t)img * 512 * 256;
#pragma unroll
  for (int r = 0; r < 8; ++r) {
    int row = m0 + r + 8 * half;
    f1r[(size_t)row * 256 + n0 + mrow] = fr[r] * inv;
    f1i[(size_t)row * 256 + n0 + mrow] = fi[r] * inv;
  }
}

// ---------------- stage C: weight + kx=127 phase reduction --------------
// One wave per (b,co,y): T = sum_{cin,x} (F1r*Wr - F1i*Wi) * e^{-i*th*127x}.

__global__ void __launch_bounds__(256) fc_stageC(
    const float* __restrict__ F1r, const float* __restrict__ F1i,
    const float* __restrict__ Wr, const float* __restrict__ Wi,
    const float* __restrict__ twc, const float* __restrict__ tws,
    float* __restrict__ Tr, float* __restrict__ Ti) {
  int wave = (blockIdx.x * blockDim.x + threadIdx.x) >> 5;  // 0..16287
  int lane = threadIdx.x & 31;
  int y = wave % 509;
  int t = wave / 509;  // b*16 + co
  int co = t & 15;
  int b = t >> 4;
  float sr = 0.0f, si = 0.0f;
  for (int cin = 0; cin < 8; ++cin) {
    const float* fr = F1r + ((size_t)(b * 8 + cin) * 512 + y) * 256;
    const float* fi = F1i + ((size_t)(b * 8 + cin) * 512 + y) * 256;
    const float* wr = Wr + ((size_t)(co * 8 + cin) * 509 + y) * 255;
    const float* wi = Wi + ((size_t)(co * 8 + cin) * 509 + y) * 255;
    for (int x = lane; x < 255; x += 32) {
      float R = fr[x] * wr[x] - fi[x] * wi[x];
      int p = (127 * x) % 509;
      sr += R * twc[p];
      si -= R * tws[p];
    }
  }
#pragma unroll
  for (int off = 16; off > 0; off >>= 1) {
    sr += __shfl_xor(sr, off, 32);
    si += __shfl_xor(si, off, 32);
  }
  if (lane == 0) {
    Tr[(size_t)t * 512 + y] = sr;
    Ti[(size_t)t * 512 + y] = si;
  }
}

// ---------------- stage D: final 509-pt DFT over y + bias ---------------
// One wave per (b,co,k), k=0..254 -> ky = 127+k.

__global__ void __launch_bounds__(256) fc_stageD(
    const float* __restrict__ Tr, const float* __restrict__ Ti,
    const float* __restrict__ bias, const float* __restrict__ twc,
    const float* __restrict__ tws, float* __restrict__ out) {
  int wave = (blockIdx.x * blockDim.x + threadIdx.x) >> 5;  // 0..8159
  int lane = threadIdx.x & 31;
  int k = wave % 255;
  int t = wave / 255;  // b*16 + co
  int co = t & 15;
  int ky = 127 + k;
  const float* tr = Tr + (size_t)t * 512;
  const float* ti = Ti + (size_t)t * 512;
  float sr = 0.0f, si = 0.0f;
  for (int y = lane; y < 509; y += 32) {
    int p = (ky * y) % 509;
    float c = twc[p], s = tws[p];
    float a = tr[y], d = ti[y];
    sr += a * c + d * s;  // Re(T * e^{-i*th*ky*y})
    si += d * c - a * s;  // Im(...)
  }
#pragma unroll
  for (int off = 16; off > 0; off >>= 1) {
    sr += __shfl_xor(sr, off, 32);
    si += __shfl_xor(si, off, 32);
  }
  if (lane == 0) {
    const float inv = 1.0f / 509.0f;  // ortho norm of second rfft2
    // out layout: (2, B, Cout, 255, 1); bias adds to real part only.
    out[(size_t)t * 255 + k] = sr * inv + bias[co];
    out[(size_t)(32 + t) * 255 + k] = si * inv;
  }
}

// ---------------- host launcher -----------------------------------------

extern "C" void kernel_launch(void* const* d_in, const int* in_sizes, int n_in,
                              void* d_out, int out_size, void* d_ws,
                              size_t ws_size, hipStream_t stream) {
  (void)in_sizes;
  (void)n_in;
  (void)out_size;
  (void)ws_size;
  const float* im = (const float*)d_in[0];    // (2,8,255,255)
  const float* Wr = (const float*)d_in[1];    // (16,8,509,255)
  const float* Wi = (const float*)d_in[2];    // (16,8,509,255)
  const float* bias = (const float*)d_in[3];  // (16,1,1)
  float* out = (float*)d_out;                 // (2,2,16,255,1)

  float* ws = (float*)d_ws;
  size_t o = 0;
  float* twc = ws + o;  o += 512;
  float* tws_ = ws + o; o += 512;
  float* E1r = ws + o;  o += 256 * 256;
  float* E1i = ws + o;  o += 256 * 256;
  float* cosT = ws + o; o += 512 * 256;
  float* sinP = ws + o; o += 512 * 256;
  float* sinN = ws + o; o += 512 * 256;
  float* imP = ws + o;  o += 4096 * 256;
  float* Gr = ws + o;   o += 4096 * 256;
  float* Gi = ws + o;   o += 4096 * 256;
  float* F1r = ws + o;  o += 16 * 512 * 256;
  float* F1i = ws + o;  o += 16 * 512 * 256;
  float* Tr = ws + o;   o += 32 * 512;
  float* Ti = ws + o;   o += 32 * 512;
  // total ~7.9M floats (~31 MB) of workspace

  fc_init_tw<<<2, 256, 0, stream>>>(twc, tws_);
  fc_init_E<<<512, 256, 0, stream>>>(twc, tws_, E1r, E1i, cosT, sinP, sinN);
  fc_pack_im<<<4096, 256, 0, stream>>>(im, imP, Gr, Gi);
  fc_stageA<<<510, 256, 0, stream>>>(imP, E1r, E1i, Gr, Gi);    // 4080 waves
  fc_stageB<<<1024, 256, 0, stream>>>(cosT, sinP, sinN, Gr, Gi, F1r, F1i);
  fc_stageC<<<2036, 256, 0, stream>>>(F1r, F1i, Wr, Wi, twc, tws_, Tr, Ti);
  fc_stageD<<<1020, 256, 0, stream>>>(Tr, Ti, bias, twc, tws_, out);
}